// Backbone_76038101008955
// MI455X (gfx1250) — compile-verified
//
#include <hip/hip_runtime.h>

// ============================================================================
// PointNet++ MSG backbone for MI455X (gfx1250, wave32).
//
//  * FPS: one workgroup per batch, coords + running min-dists in registers,
//    LDS tree argmax (serial bottleneck, latency-optimized).
//  * Shared-MLP layers: Y = relu(X @ W^T + b) on V_WMMA_F32_16X16X4_F32
//    (fp32 A/B/C). Ragged-K weights pre-padded -> branch-free inner loop.
//    Each wave computes an MT x NT grid of 16x16 tiles: loads are batched
//    ahead of the WMMA burst so the scheduler can cover them with one wait
//    and keep MT*NT independent accumulation chains in flight.
//  * Ball query: wave32 __ballot + prefix popcount = "first k ascending
//    indices in ball, pad with first hit" (matches reference top_k trick).
// ============================================================================

typedef __attribute__((ext_vector_type(2))) float v2f;
typedef __attribute__((ext_vector_type(8))) float v8f;

// ---------------------------------------------------------------------------
// Farthest point sampling: one block per batch. N = BLOCK*PPT points.
// ---------------------------------------------------------------------------
template <int BLOCK, int PPT, int NPOINT>
__global__ __launch_bounds__(BLOCK) void fps_kernel(const float* __restrict__ xyz,
                                                    int* __restrict__ out) {
  const int b = blockIdx.x;
  const int t = threadIdx.x;
  const float* px = xyz + (size_t)b * (BLOCK * PPT) * 3;

  float lx[PPT], ly[PPT], lz[PPT], ld[PPT];
#pragma unroll
  for (int i = 0; i < PPT; ++i) {
    const int p = t + i * BLOCK;
    lx[i] = px[p * 3 + 0];
    ly[i] = px[p * 3 + 1];
    lz[i] = px[p * 3 + 2];
    ld[i] = 1e10f;
  }

  __shared__ float sval[BLOCK];
  __shared__ int sidx[BLOCK];
  __shared__ float cpt[3];

  if (t == 0) {
    out[b * NPOINT] = 0;  // first sample is index 0 (matches reference)
    cpt[0] = px[0];
    cpt[1] = px[1];
    cpt[2] = px[2];
  }
  __syncthreads();

  for (int j = 1; j < NPOINT; ++j) {
    const float cx = cpt[0], cy = cpt[1], cz = cpt[2];
    float best = -1.0f;
    int bidx = 0;
#pragma unroll
    for (int i = 0; i < PPT; ++i) {
      const float dx = lx[i] - cx, dy = ly[i] - cy, dz = lz[i] - cz;
      const float d = dx * dx + dy * dy + dz * dz;
      ld[i] = fminf(ld[i], d);
      if (ld[i] > best) {
        best = ld[i];
        bidx = t + i * BLOCK;
      }
    }
    sval[t] = best;
    sidx[t] = bidx;
    __syncthreads();
#pragma unroll
    for (int s = BLOCK / 2; s > 0; s >>= 1) {
      if (t < s) {
        const float ov = sval[t + s];
        const int oi = sidx[t + s];
        if (ov > sval[t] || (ov == sval[t] && oi < sidx[t])) {
          sval[t] = ov;
          sidx[t] = oi;
        }
      }
      __syncthreads();
    }
    if (t == 0) {
      const int bi = sidx[0];
      out[b * NPOINT + j] = bi;
      cpt[0] = px[bi * 3 + 0];
      cpt[1] = px[bi * 3 + 1];
      cpt[2] = px[bi * 3 + 2];
    }
    __syncthreads();
  }
}

// ---------------------------------------------------------------------------
// Gather sampled centers: dst[b,s,:] = src[b, idx[b,s], :]
// ---------------------------------------------------------------------------
__global__ void gather_points_kernel(const float* __restrict__ src,
                                     const int* __restrict__ idx,
                                     float* __restrict__ dst, int Npts, int S,
                                     int total) {
  const int t = blockIdx.x * blockDim.x + threadIdx.x;
  if (t >= total) return;
  const int b = t / S;
  const int i = idx[t];
  const float* p = src + ((size_t)b * Npts + i) * 3;
  dst[t * 3 + 0] = p[0];
  dst[t * 3 + 1] = p[1];
  dst[t * 3 + 2] = p[2];
}

// ---------------------------------------------------------------------------
// Ball query: one wave32 per center.
// ---------------------------------------------------------------------------
__global__ __launch_bounds__(256) void ball_query_kernel(
    const float* __restrict__ xyz, const float* __restrict__ centers,
    int* __restrict__ out, int Npts, int S, int K, float r2, int total) {
  const int lane = threadIdx.x & 31;
  const int wid = (blockIdx.x * blockDim.x + threadIdx.x) >> 5;
  if (wid >= total) return;
  const int b = wid / S;
  const float cx = centers[wid * 3 + 0];
  const float cy = centers[wid * 3 + 1];
  const float cz = centers[wid * 3 + 2];
  const float* px = xyz + (size_t)b * Npts * 3;
  int* o = out + (size_t)wid * K;

  int cnt = 0;
  for (int base = 0; base < Npts && cnt < K; base += 32) {
    const int p = base + lane;  // Npts is a multiple of 32
    const float dx = px[p * 3 + 0] - cx;
    const float dy = px[p * 3 + 1] - cy;
    const float dz = px[p * 3 + 2] - cz;
    const bool hit = (dx * dx + dy * dy + dz * dz) < r2;
    const unsigned mask = (unsigned)__ballot(hit);
    if (hit) {
      const int pos = cnt + __popc(mask & ((1u << lane) - 1u));
      if (pos < K) o[pos] = p;
    }
    cnt += __popc(mask);
  }
  if (cnt > K) cnt = K;
  if (lane == 0) {
    const int first = o[0];
    for (int q = cnt; q < K; ++q) o[q] = first;  // pad with first hit
  }
}

// ---------------------------------------------------------------------------
// Build grouped MLP input rows: X[row, 0:3] = xyz[p]-center, then feats,
// zero pad to Kpad. row = (b*S+s)*K + k. One block per row.
// ---------------------------------------------------------------------------
__global__ void group_build_kernel(const float* __restrict__ xyz,
                                   const float* __restrict__ centers,
                                   const float* __restrict__ feats,
                                   const int* __restrict__ ballidx,
                                   float* __restrict__ X, int Nsrc, int S,
                                   int K, int Cf, int Kpad) {
  const long row = blockIdx.x;
  const long bs = row / K;
  const int b = (int)(bs / S);
  const int p = ballidx[row];
  const float* pc = centers + bs * 3;
  const float* pp = xyz + ((long)b * Nsrc + p) * 3;
  float* xr = X + row * Kpad;
  for (int c = threadIdx.x; c < Kpad; c += blockDim.x) {
    float v;
    if (c < 3)
      v = pp[c] - pc[c];
    else if (c < 3 + Cf)
      v = feats[((long)b * Nsrc + p) * Cf + (c - 3)];
    else
      v = 0.0f;
    xr[c] = v;
  }
}

// ---------------------------------------------------------------------------
// Zero-pad weights W[N,K] -> Wp[N,Kpad] (run only for ragged-K first layers).
// ---------------------------------------------------------------------------
__global__ void pad_weights_kernel(const float* __restrict__ W,
                                   float* __restrict__ Wp, int N, int K,
                                   int Kpad) {
  const int t = blockIdx.x * blockDim.x + threadIdx.x;
  if (t >= N * Kpad) return;
  const int n = t / Kpad;
  const int k = t % Kpad;
  Wp[t] = (k < K) ? W[n * K + k] : 0.0f;
}

// ---------------------------------------------------------------------------
// WMMA fp32 GEMM: Y[M,N] = relu(X[M,Kpad] @ Wp[N,Kpad]^T + bias[N]).
// Each wave computes MT x NT 16x16 tiles. Per k-step: MT A-frag loads +
// NT B-frag loads are issued as a batch, then MT*NT WMMAs run back-to-back.
//   A 16x4 frag: lanes 0-15 -> K {k0,k0+1}, lanes 16-31 -> K {k0+2,k0+3}
//   B 4x16 frag: lane&15 = column, same K split
//   D frag:      VGPR r -> row r (lanes 0-15) / r+8 (lanes 16-31), col lane&15
// ---------------------------------------------------------------------------
template <int MT, int NT>
__global__ __launch_bounds__(128) void gemm_bias_relu_f32(
    const float* __restrict__ X, const float* __restrict__ Wp,
    const float* __restrict__ bias, float* __restrict__ Y, int M, int N,
    int Kpad) {
  const int lane = threadIdx.x & 31;
  const int wave = threadIdx.x >> 5;
  const int mgroups = (M >> 4) / MT;
  const int ngroups = (N >> 4) / NT;
  const int tile = blockIdx.x * 4 + wave;
  if (tile >= mgroups * ngroups) return;
  const int mg = tile / ngroups;
  const int ng = tile % ngroups;

  const int half = lane >> 4;  // which K pair this lane holds
  const int l15 = lane & 15;

  const float* xp[MT];
  const float* wp[NT];
  float bval[NT];
  int bcol[NT];
  v8f acc[MT][NT];
#pragma unroll
  for (int i = 0; i < MT; ++i) {
    xp[i] = X + (long)((mg * MT + i) * 16 + l15) * Kpad + half * 2;
#pragma unroll
    for (int j = 0; j < NT; ++j) acc[i][j] = (v8f){};
  }
#pragma unroll
  for (int j = 0; j < NT; ++j) {
    bcol[j] = (ng * NT + j) * 16 + l15;
    wp[j] = Wp + (long)bcol[j] * Kpad + half * 2;
    bval[j] = bias[bcol[j]];
  }

#pragma unroll 2
  for (int k0 = 0; k0 < Kpad; k0 += 4) {
    // Batch all independent fragment loads first ...
    v2f a[MT], bf[NT];
#pragma unroll
    for (int i = 0; i < MT; ++i) a[i] = *(const v2f*)(xp[i] + k0);
#pragma unroll
    for (int j = 0; j < NT; ++j) bf[j] = *(const v2f*)(wp[j] + k0);
    // ... then run the whole WMMA burst from registers.
#pragma unroll
    for (int i = 0; i < MT; ++i)
#pragma unroll
      for (int j = 0; j < NT; ++j)
        acc[i][j] = __builtin_amdgcn_wmma_f32_16x16x4_f32(
            false, a[i], false, bf[j], (short)0, acc[i][j], false, false);
  }

#pragma unroll
  for (int i = 0; i < MT; ++i) {
    const int rbase = (mg * MT + i) * 16 + half * 8;
#pragma unroll
    for (int j = 0; j < NT; ++j) {
#pragma unroll
      for (int r = 0; r < 8; ++r) {
        float v = acc[i][j][r] + bval[j];
        v = v > 0.0f ? v : 0.0f;  // ReLU on every layer (matches _mlp)
        Y[(long)(rbase + r) * N + bcol[j]] = v;
      }
    }
  }
}

// ---------------------------------------------------------------------------
// Max-pool over K samples, write into concatenated feature buffer at coff.
// ---------------------------------------------------------------------------
__global__ void maxpool_concat_kernel(const float* __restrict__ Y,
                                      float* __restrict__ outf, int K, int C,
                                      int Ctot, int coff, long total) {
  const long t = blockIdx.x * (long)blockDim.x + threadIdx.x;
  if (t >= total) return;
  const long bs = t / C;
  const int c = (int)(t % C);
  const float* y = Y + (bs * K) * (long)C + c;
  float m = y[0];
  for (int j = 1; j < K; ++j) m = fmaxf(m, y[(long)j * C]);
  outf[bs * (long)Ctot + coff + c] = m;
}

// ---------------------------------------------------------------------------
// 3-NN (squared distances, ties keep lower index) + inverse-distance weights.
// ---------------------------------------------------------------------------
__global__ void three_nn_kernel(const float* __restrict__ unk,
                                const float* __restrict__ known,
                                int* __restrict__ oidx, float* __restrict__ ow,
                                int Nd, int Ns, int total) {
  const int t = blockIdx.x * blockDim.x + threadIdx.x;
  if (t >= total) return;
  const int b = t / Nd;
  const float ux = unk[t * 3 + 0], uy = unk[t * 3 + 1], uz = unk[t * 3 + 2];
  const float* kp = known + (long)b * Ns * 3;
  float d0 = 3e38f, d1 = 3e38f, d2 = 3e38f;
  int i0 = 0, i1 = 0, i2 = 0;
  for (int j = 0; j < Ns; ++j) {
    const float dx = kp[j * 3 + 0] - ux;
    const float dy = kp[j * 3 + 1] - uy;
    const float dz = kp[j * 3 + 2] - uz;
    const float d = dx * dx + dy * dy + dz * dz;
    if (d < d0) {
      d2 = d1; i2 = i1;
      d1 = d0; i1 = i0;
      d0 = d;  i0 = j;
    } else if (d < d1) {
      d2 = d1; i2 = i1;
      d1 = d;  i1 = j;
    } else if (d < d2) {
      d2 = d;  i2 = j;
    }
  }
  float w0 = 1.0f / (d0 + 1e-8f);
  float w1 = 1.0f / (d1 + 1e-8f);
  float w2 = 1.0f / (d2 + 1e-8f);
  const float s = w0 + w1 + w2;
  ow[t * 3 + 0] = w0 / s;
  ow[t * 3 + 1] = w1 / s;
  ow[t * 3 + 2] = w2 / s;
  oidx[t * 3 + 0] = i0;
  oidx[t * 3 + 1] = i1;
  oidx[t * 3 + 2] = i2;
}

// ---------------------------------------------------------------------------
// Build FP MLP input: X[u, 0:C] = sum_i w_i * kf[idx_i], X[u, C:C+Cs] = skip.
// ---------------------------------------------------------------------------
__global__ __launch_bounds__(256) void interp_build_kernel(
    const int* __restrict__ knnidx, const float* __restrict__ knnw,
    const float* __restrict__ kf, const float* __restrict__ skip,
    float* __restrict__ X, int Nd, int Ns, int C, int Cs) {
  const long u = blockIdx.x;
  const int b = (int)(u / Nd);
  const int i0 = knnidx[u * 3 + 0], i1 = knnidx[u * 3 + 1],
            i2 = knnidx[u * 3 + 2];
  const float w0 = knnw[u * 3 + 0], w1 = knnw[u * 3 + 1], w2 = knnw[u * 3 + 2];
  const float* f0 = kf + ((long)b * Ns + i0) * C;
  const float* f1 = kf + ((long)b * Ns + i1) * C;
  const float* f2 = kf + ((long)b * Ns + i2) * C;
  const int Ct = C + Cs;
  float* xr = X + u * (long)Ct;
  for (int c = threadIdx.x; c < Ct; c += blockDim.x) {
    xr[c] = (c < C) ? (w0 * f0[c] + w1 * f1[c] + w2 * f2[c])
                    : skip[u * (long)Cs + (c - C)];
  }
}

// ============================================================================
// Host orchestration
// ============================================================================
extern "C" void kernel_launch(void* const* d_in, const int* in_sizes, int n_in,
                              void* d_out, int out_size, void* d_ws,
                              size_t ws_size, hipStream_t stream) {
  constexpr int B = 4;
  constexpr int N0 = 8192, S1 = 2048, S2 = 512, S3 = 128;

  // ---- Input flattening order detection -----------------------------------
  // Order A (jax tree_leaves, dict keys sorted): fp(12) + sa(36) + xyz.
  // Order B (insertion order): xyz + sa(36) + fp(12).
  int fpBase, saBase, xyzIdx;
  if (in_sizes[0] == B * N0 * 3) {  // xyz first -> order B
    xyzIdx = 0;
    saBase = 1;
    fpBase = 37;
  } else {  // order A
    fpBase = 0;
    saBase = 12;
    xyzIdx = n_in - 1;
  }
  const float* xyz0 = (const float*)d_in[xyzIdx];
  auto saW = [&](int L, int s, int l) {
    return (const float*)d_in[saBase + L * 12 + s * 6 + l * 2];
  };
  auto saBp = [&](int L, int s, int l) {
    return (const float*)d_in[saBase + L * 12 + s * 6 + l * 2 + 1];
  };
  auto fpW = [&](int m, int l) {
    return (const float*)d_in[fpBase + m * 4 + l * 2];
  };
  auto fpBp = [&](int m, int l) {
    return (const float*)d_in[fpBase + m * 4 + l * 2 + 1];
  };

  // ---- Workspace bump allocator (assumes ws_size >= ~160 MB) --------------
  char* wsp = (char*)d_ws;
  size_t off = 0;
  auto alloc = [&](size_t bytes) -> void* {
    off = (off + 255) & ~(size_t)255;
    void* p = wsp + off;
    off += bytes;
    return p;
  };
  int* fpsidx1 = (int*)alloc(sizeof(int) * B * S1);
  int* fpsidx2 = (int*)alloc(sizeof(int) * B * S2);
  int* fpsidx3 = (int*)alloc(sizeof(int) * B * S3);
  float* cxyz1 = (float*)alloc(sizeof(float) * B * S1 * 3);
  float* cxyz2 = (float*)alloc(sizeof(float) * B * S2 * 3);
  float* cxyz3 = (float*)alloc(sizeof(float) * B * S3 * 3);
  int* ball = (int*)alloc(sizeof(int) * B * S1 * 32);  // reused per scale
  float* wpad = (float*)alloc(sizeof(float) * 65536);  // padded ragged-K W
  float* feats1 = (float*)alloc(sizeof(float) * B * S1 * 96);
  float* feats2 = (float*)alloc(sizeof(float) * B * S2 * 256);
  float* feats3 = (float*)alloc(sizeof(float) * B * S3 * 1024);
  float* fpout2 = (float*)alloc(sizeof(float) * B * S2 * 256);
  float* fpout1 = (float*)alloc(sizeof(float) * B * S1 * 128);
  int* knnidx = (int*)alloc(sizeof(int) * B * N0 * 3);
  float* knnw = (float*)alloc(sizeof(float) * B * N0 * 3);
  float* pingA = (float*)alloc(sizeof(float) * (size_t)16777216);  // 64 MB
  float* pingB = (float*)alloc(sizeof(float) * (size_t)16777216);  // 64 MB
  (void)ws_size;
  (void)out_size;

  // ---- GEMM launcher (pads ragged-K weights, dispatches MT/NT variant) ----
  auto run_gemm = [&](const float* X, const float* W, const float* bb,
                      float* Y, long M, int N, int K, int Kpad) {
    const float* Wuse = W;
    if (K != Kpad) {
      const int tot = N * Kpad;
      pad_weights_kernel<<<(tot + 255) / 256, 256, 0, stream>>>(W, wpad, N, K,
                                                                Kpad);
      Wuse = wpad;
    }
    const int mtiles = (int)(M / 16);
    const int ntiles = N / 16;
    const int MT = (mtiles % 2 == 0) ? 2 : 1;
    const int NT = ((ntiles & 3) == 0) ? 4 : (((ntiles & 1) == 0) ? 2 : 1);
    const long tiles = (long)(mtiles / MT) * (ntiles / NT);
    const int blocks = (int)((tiles + 3) / 4);
    if (MT == 2 && NT == 4)
      gemm_bias_relu_f32<2, 4><<<blocks, 128, 0, stream>>>(X, Wuse, bb, Y,
                                                           (int)M, N, Kpad);
    else if (MT == 2 && NT == 2)
      gemm_bias_relu_f32<2, 2><<<blocks, 128, 0, stream>>>(X, Wuse, bb, Y,
                                                           (int)M, N, Kpad);
    else if (MT == 2 && NT == 1)
      gemm_bias_relu_f32<2, 1><<<blocks, 128, 0, stream>>>(X, Wuse, bb, Y,
                                                           (int)M, N, Kpad);
    else if (MT == 1 && NT == 4)
      gemm_bias_relu_f32<1, 4><<<blocks, 128, 0, stream>>>(X, Wuse, bb, Y,
                                                           (int)M, N, Kpad);
    else if (MT == 1 && NT == 2)
      gemm_bias_relu_f32<1, 2><<<blocks, 128, 0, stream>>>(X, Wuse, bb, Y,
                                                           (int)M, N, Kpad);
    else
      gemm_bias_relu_f32<1, 1><<<blocks, 128, 0, stream>>>(X, Wuse, bb, Y,
                                                           (int)M, N, Kpad);
  };

  // ---- One SA scale: ball query -> group -> 3-layer MLP -> maxpool --------
  auto run_sa_scale = [&](const float* src_xyz, const float* src_feats,
                          int Nsrc, int Cf, const float* cxyz, int S, int K,
                          float radius, int L, int sc, const int* dims,
                          int Kpad0, float* featout, int Ctot, int coff) {
    const int total = B * S;
    ball_query_kernel<<<(total + 7) / 8, 256, 0, stream>>>(
        src_xyz, cxyz, ball, Nsrc, S, K, radius * radius, total);
    const long rows = (long)total * K;
    const int gb = (Kpad0 <= 32) ? 32 : ((Kpad0 <= 128) ? 128 : 256);
    group_build_kernel<<<(int)rows, gb, 0, stream>>>(
        src_xyz, cxyz, src_feats, ball, pingA, Nsrc, S, K, Cf, Kpad0);
    const float* cur = pingA;
    float* nxt = pingB;
    int kc = dims[0];
    int kpad = Kpad0;
    for (int l = 0; l < 3; ++l) {
      const int n = dims[l + 1];
      run_gemm(cur, saW(L, sc, l), saBp(L, sc, l), nxt, rows, n, kc, kpad);
      cur = nxt;
      nxt = (nxt == pingB) ? pingA : pingB;
      kc = n;
      kpad = n;
    }
    const long tbsc = (long)total * kc;
    maxpool_concat_kernel<<<(int)((tbsc + 255) / 256), 256, 0, stream>>>(
        cur, featout, K, kc, Ctot, coff, tbsc);
  };

  // ---- One FP level: 3-NN -> interpolate(+skip concat) -> 2-layer MLP -----
  auto run_fp = [&](const float* unk_xyz, int Nd, const float* known_xyz,
                    int Ns, const float* kf, int C, const float* skip, int Cs,
                    int m, const int* dims, float* outbuf) {
    const int total = B * Nd;
    three_nn_kernel<<<(total + 255) / 256, 256, 0, stream>>>(
        unk_xyz, known_xyz, knnidx, knnw, Nd, Ns, total);
    interp_build_kernel<<<total, 256, 0, stream>>>(knnidx, knnw, kf, skip,
                                                   pingA, Nd, Ns, C, Cs);
    run_gemm(pingA, fpW(m, 0), fpBp(m, 0), pingB, total, dims[1], dims[0],
             dims[0]);
    run_gemm(pingB, fpW(m, 1), fpBp(m, 1), outbuf, total, dims[2], dims[1],
             dims[1]);
  };

  // ======================= Set abstraction =================================
  // Level 1: 8192 -> 2048 centers, feats 96 = 32 + 64
  fps_kernel<1024, 8, S1><<<B, 1024, 0, stream>>>(xyz0, fpsidx1);
  gather_points_kernel<<<(B * S1 + 255) / 256, 256, 0, stream>>>(
      xyz0, fpsidx1, cxyz1, N0, S1, B * S1);
  {
    const int d0[4] = {3, 16, 16, 32};
    run_sa_scale(xyz0, nullptr, N0, 0, cxyz1, S1, 16, 0.01f, 0, 0, d0, 4,
                 feats1, 96, 0);
    const int d1[4] = {3, 32, 32, 64};
    run_sa_scale(xyz0, nullptr, N0, 0, cxyz1, S1, 32, 0.03f, 0, 1, d1, 4,
                 feats1, 96, 32);
  }

  // Level 2: 2048 -> 512 centers, feats 256 = 128 + 128 (Cin 99 -> pad 100)
  fps_kernel<1024, 2, S2><<<B, 1024, 0, stream>>>(cxyz1, fpsidx2);
  gather_points_kernel<<<(B * S2 + 255) / 256, 256, 0, stream>>>(
      cxyz1, fpsidx2, cxyz2, S1, S2, B * S2);
  {
    const int d0[4] = {99, 64, 64, 128};
    run_sa_scale(cxyz1, feats1, S1, 96, cxyz2, S2, 16, 0.025f, 1, 0, d0, 100,
                 feats2, 256, 0);
    const int d1[4] = {99, 64, 96, 128};
    run_sa_scale(cxyz1, feats1, S1, 96, cxyz2, S2, 32, 0.05f, 1, 1, d1, 100,
                 feats2, 256, 128);
  }

  // Level 3: 512 -> 128 centers, feats 1024 = 512 + 512 (Cin 259 -> pad 260)
  fps_kernel<512, 1, S3><<<B, 512, 0, stream>>>(cxyz2, fpsidx3);
  gather_points_kernel<<<(B * S3 + 255) / 256, 256, 0, stream>>>(
      cxyz2, fpsidx3, cxyz3, S2, S3, B * S3);
  {
    const int d0[4] = {259, 128, 256, 512};
    run_sa_scale(cxyz2, feats2, S2, 256, cxyz3, S3, 16, 0.1f, 2, 0, d0, 260,
                 feats3, 1024, 0);
    run_sa_scale(cxyz2, feats2, S2, 256, cxyz3, S3, 32, 0.15f, 2, 1, d0, 260,
                 feats3, 1024, 512);
  }

  // ======================= Feature propagation =============================
  {
    const int d2[3] = {1280, 256, 256};  // params['fp'][-1]
    run_fp(cxyz2, S2, cxyz3, S3, feats3, 1024, feats2, 256, 2, d2, fpout2);
    const int d1[3] = {352, 256, 128};  // params['fp'][-2]
    run_fp(cxyz1, S1, cxyz2, S2, fpout2, 256, feats1, 96, 1, d1, fpout1);
    const int d0[3] = {128, 128, 128};  // params['fp'][-3], no skip feats
    run_fp(xyz0, N0, cxyz1, S1, fpout1, 128, nullptr, 0, 0, d0,
           (float*)d_out);
  }
}